// HomoGCL_56745107915246
// MI455X (gfx1250) — compile-verified
//
#include <hip/hip_runtime.h>

typedef __attribute__((ext_vector_type(2))) float v2f;
typedef __attribute__((ext_vector_type(8))) float v8f;

#define IN_DIM 512
#define HID    128

// ---------------- utility kernels ----------------

__global__ void zero_f4(float4* __restrict__ p, int nvec) {
  int i = blockIdx.x * blockDim.x + threadIdx.x;
  if (i < nvec) p[i] = make_float4(0.f, 0.f, 0.f, 0.f);
}

__global__ void degree_kernel(const int* __restrict__ src, const int* __restrict__ dst,
                              float* __restrict__ deg_out, float* __restrict__ deg_in, int E) {
  int e = blockIdx.x * blockDim.x + threadIdx.x;
  if (e < E) {
    atomicAdd(&deg_out[src[e]], 1.0f);
    atomicAdd(&deg_in[dst[e]], 1.0f);
  }
}

__global__ void rsqrt_kernel(float* __restrict__ d, int n) {
  int i = blockIdx.x * blockDim.x + threadIdx.x;
  if (i < n) d[i] = rsqrtf(fmaxf(d[i], 1.0f));
}

// Pack W[K][128] into WMMA B-fragment order so each lane's (k, k+1) pair is
// contiguous: Wp[(k>>2)*2 + ((k>>1)&1)][n][k&1]. One coalesced b64 per step.
__global__ void pack_W(const float* __restrict__ W, float* __restrict__ Wp, int K) {
  int i = blockIdx.x * blockDim.x + threadIdx.x;
  if (i < K * HID) {
    int k = i >> 7, n = i & 127;
    int grp = k >> 2, half = (k >> 1) & 1, lo = k & 1;
    Wp[(((grp << 1) + half) << 8) + (n << 1) + lo] = W[i];
  }
}

// ---------------- fp32 WMMA GEMM:  D = rowscale * (X[N,K] @ W[K,128]) ----------------
// Block = 256 threads = 8 waves, covers a 32-row M-tile; wave w covers cols [16w,16w+16)
// with TWO v8f accumulators (rows m0..+15 and m0+16..+31) sharing each B fragment.
// Bounds handling is hoisted to block granularity: only the single tail block
// (N % 32 != 0) takes the guarded path, so 1562/1563 blocks are branch-free.

template <int K>
__global__ __launch_bounds__(256) void gemm_wmma(const float* __restrict__ X,
                                                 const float* __restrict__ Wp,
                                                 const float* __restrict__ rowscale,
                                                 float* __restrict__ D, int N) {
  __shared__ float ldsX[32 * K];

  const int m0 = blockIdx.x * 32;
  const bool full = (m0 + 32 <= N);
  const int rowvec = K / 4;                 // float4s per row

  if (full) {
    // tile rows are contiguous: one linear float4 copy of 32*K floats
    const float4* __restrict__ Xt = (const float4*)(X + (size_t)m0 * K);
    float4* L4 = (float4*)ldsX;
    #pragma unroll 4
    for (int i = threadIdx.x; i < 32 * rowvec; i += 256) L4[i] = Xt[i];
  } else {
    for (int i = threadIdx.x; i < 32 * rowvec; i += 256) {
      int row = i / rowvec;                 // constexpr power-of-two divide
      float4 v = make_float4(0.f, 0.f, 0.f, 0.f);
      if (m0 + row < N)
        v = ((const float4*)(X + (size_t)(m0 + row) * K))[i - row * rowvec];
      ((float4*)ldsX)[i] = v;
    }
  }
  __syncthreads();

  const int wave = threadIdx.x >> 5;
  const int lane = threadIdx.x & 31;
  const int m    = lane & 15;               // A row within 16-tile / output col within wave
  const int half = lane >> 4;               // 0: lanes 0-15, 1: lanes 16-31
  const int n    = wave * 16 + m;           // output column (0..127)

  v8f acc0 = {};
  v8f acc1 = {};
  const v2f* __restrict__ Bp = (const v2f*)Wp;

  #pragma unroll 4
  for (int kk = 0; kk < K; kk += 4) {
    const int kb = kk + half * 2;
    v2f b = Bp[((kk >> 1) + half) * 128 + n];      // one global_load_b64, coalesced
    v2f a0, a1;
    a0.x = ldsX[m * K + kb];
    a0.y = ldsX[m * K + kb + 1];
    a1.x = ldsX[(m + 16) * K + kb];
    a1.y = ldsX[(m + 16) * K + kb + 1];
    acc0 = __builtin_amdgcn_wmma_f32_16x16x4_f32(false, a0, false, b, (short)0, acc0, false, false);
    acc1 = __builtin_amdgcn_wmma_f32_16x16x4_f32(false, a1, false, b, (short)0, acc1, false, false);
  }

  // epilogue: scale rows by inv_sqrt_out
  if (full) {
    // rowscale for this lane's 8+8 rows: four b128 loads (16B-aligned offsets)
    const float4* rs0 = (const float4*)(rowscale + m0 + half * 8);
    const float4* rs1 = (const float4*)(rowscale + m0 + 16 + half * 8);
    float4 sa = rs0[0], sb = rs0[1];
    float4 sc = rs1[0], sd = rs1[1];
    float s0[8] = {sa.x, sa.y, sa.z, sa.w, sb.x, sb.y, sb.z, sb.w};
    float s1[8] = {sc.x, sc.y, sc.z, sc.w, sd.x, sd.y, sd.z, sd.w};
    float* __restrict__ D0 = D + (size_t)(m0 + half * 8) * HID + n;
    float* __restrict__ D1 = D + (size_t)(m0 + 16 + half * 8) * HID + n;
    #pragma unroll
    for (int j = 0; j < 8; ++j) D0[j * HID] = acc0[j] * s0[j];
    #pragma unroll
    for (int j = 0; j < 8; ++j) D1[j * HID] = acc1[j] * s1[j];
  } else {
    #pragma unroll
    for (int j = 0; j < 8; ++j) {
      int r0 = m0 + half * 8 + j;
      if (r0 < N) D[(size_t)r0 * HID + n] = acc0[j] * rowscale[r0];
    }
    #pragma unroll
    for (int j = 0; j < 8; ++j) {
      int r1 = m0 + 16 + half * 8 + j;
      if (r1 < N) D[(size_t)r1 * HID + n] = acc1[j] * rowscale[r1];
    }
  }
}

// ---------------- edge aggregation: agg[dst] += h[src]  (128 floats/edge) ----------------
// One wave per edge; each lane moves a float4 (32 lanes * 16B = 512B = full row).
// h and agg are L2-resident (25.6 MB each).

__global__ __launch_bounds__(256) void aggregate_kernel(const float* __restrict__ h,
                                                        const int* __restrict__ src,
                                                        const int* __restrict__ dst,
                                                        float* __restrict__ agg, int E) {
  const int e    = blockIdx.x * 8 + (threadIdx.x >> 5);
  const int lane = threadIdx.x & 31;
  if (e < E) {
    const int s = src[e];
    const int d = dst[e];
    float4 v = ((const float4*)(h + (size_t)s * HID))[lane];
    float* out = agg + (size_t)d * HID + lane * 4;
    atomicAdd(out + 0, v.x);
    atomicAdd(out + 1, v.y);
    atomicAdd(out + 2, v.z);
    atomicAdd(out + 3, v.w);
  }
}

// ---------------- epilogue: out = (agg * inv_in[node]) + bias[c], optional ReLU ----------------

__global__ void finish_kernel(const float4* __restrict__ agg, const float* __restrict__ inv_in,
                              const float4* __restrict__ bias, float4* __restrict__ out,
                              int nvec, int relu) {
  int i = blockIdx.x * blockDim.x + threadIdx.x;
  if (i < nvec) {
    float s = inv_in[i >> 5];
    float4 a = agg[i];
    float4 b = bias[i & 31];
    float4 v = make_float4(a.x * s + b.x, a.y * s + b.y, a.z * s + b.z, a.w * s + b.w);
    if (relu) {
      v.x = fmaxf(v.x, 0.f); v.y = fmaxf(v.y, 0.f);
      v.z = fmaxf(v.z, 0.f); v.w = fmaxf(v.w, 0.f);
    }
    out[i] = v;
  }
}

// ---------------- host-side orchestration ----------------

extern "C" void kernel_launch(void* const* d_in, const int* in_sizes, int n_in,
                              void* d_out, int out_size, void* d_ws, size_t ws_size,
                              hipStream_t stream) {
  (void)n_in; (void)out_size; (void)ws_size;

  const float* W1 = (const float*)d_in[9];
  const float* b1 = (const float*)d_in[10];
  const float* W2 = (const float*)d_in[11];
  const float* b2 = (const float*)d_in[12];

  const int Nn = in_sizes[0] / IN_DIM;   // 50000
  const int E  = in_sizes[1];            // 800000
  const int NH = Nn * HID;               // 6.4M

  float* ws      = (float*)d_ws;
  float* inv_out = ws;                                   // [Nn]
  float* inv_in  = ws + Nn;                              // [Nn]
  float* hbuf    = ws + 2 * (size_t)Nn;                  // [Nn*128]
  float* tbuf    = hbuf + (size_t)NH;                    // [Nn*128]
  float* agg     = tbuf + (size_t)NH;                    // [Nn*128]
  float* Wp1     = agg + (size_t)NH;                     // [512*128]
  float* Wp2     = Wp1 + IN_DIM * HID;                   // [128*128]
  float* outp    = (float*)d_out;

  const int thr = 256;
  const dim3 blkE((E + thr - 1) / thr);
  const dim3 blkN2((2 * Nn + thr - 1) / thr);
  const dim3 blkN2v((2 * Nn / 4 + thr - 1) / thr);
  const dim3 blkNHv((NH / 4 + thr - 1) / thr);
  const dim3 blkGemm((Nn + 31) / 32);
  const dim3 blkAgg((E + 7) / 8);

  // one-shot: pack W1/W2 into B-fragment order
  pack_W<<<(IN_DIM * HID + thr - 1) / thr, thr, 0, stream>>>(W1, Wp1, IN_DIM);
  pack_W<<<(HID * HID + thr - 1) / thr, thr, 0, stream>>>(W2, Wp2, HID);

  for (int g = 0; g < 3; ++g) {
    const float* feat = (const float*)d_in[g * 3 + 0];
    const int*   src  = (const int*)d_in[g * 3 + 1];
    const int*   dst  = (const int*)d_in[g * 3 + 2];

    // degree -> inv-sqrt norms (inv_out/inv_in contiguous: zero both at once)
    zero_f4<<<blkN2v, thr, 0, stream>>>((float4*)inv_out, 2 * Nn / 4);
    degree_kernel<<<blkE, thr, 0, stream>>>(src, dst, inv_out, inv_in, E);
    rsqrt_kernel<<<blkN2, thr, 0, stream>>>(inv_out, 2 * Nn);

    // layer 1: h = relu( inv_in .* A.(inv_out .* (feat @ W1)) + b1 )
    gemm_wmma<IN_DIM><<<blkGemm, thr, 0, stream>>>(feat, Wp1, inv_out, hbuf, Nn);
    zero_f4<<<blkNHv, thr, 0, stream>>>((float4*)agg, NH / 4);
    aggregate_kernel<<<blkAgg, thr, 0, stream>>>(hbuf, src, dst, agg, E);
    finish_kernel<<<blkNHv, thr, 0, stream>>>((const float4*)agg, inv_in, (const float4*)b1,
                                              (float4*)hbuf, NH / 4, 1);

    // layer 2: z = inv_in .* A.(inv_out .* (h @ W2)) + b2
    gemm_wmma<HID><<<blkGemm, thr, 0, stream>>>(hbuf, Wp2, inv_out, tbuf, Nn);
    zero_f4<<<blkNHv, thr, 0, stream>>>((float4*)agg, NH / 4);
    aggregate_kernel<<<blkAgg, thr, 0, stream>>>(tbuf, src, dst, agg, E);
    finish_kernel<<<blkNHv, thr, 0, stream>>>((const float4*)agg, inv_in, (const float4*)b2,
                                              (float4*)(outp + (size_t)g * NH), NH / 4, 0);
  }
}